// FSAS_41394894799833
// MI455X (gfx1250) — compile-verified
//
#include <hip/hip_runtime.h>

typedef float v2f __attribute__((ext_vector_type(2)));
typedef float v8f __attribute__((ext_vector_type(8)));

#define HW    256
#define HW2   65536
#define CIN   64
#define QKVC  384
#define TILE  16
#define HALO  18
#define NPOS  324   // 18*18 halo positions
#define NPOSP 336   // padded to 16-multiple; pad columns are pre-zeroed
#define NT_A  21    // 336/16
#define CHUNK 16
#define PB    130   // padded row stride for [row][128ch] LDS layouts

__device__ __forceinline__ v8f wmma4(v2f a, v2f b, v8f c) {
  // V_WMMA_F32_16X16X4_F32 : D = A(16x4) * B(4x16) + C(16x16), all f32
  return __builtin_amdgcn_wmma_f32_16x16x4_f32(false, a, false, b, (short)0, c,
                                               false, false);
}

// async global->LDS scatter (GLOBAL_LOAD_ASYNC_TO_LDS_B32, GVS mode):
// copies one dword from global memory at saddr plus global_byte_off to the LDS byte address lds_byte_addr; tracked by ASYNCcnt.
__device__ __forceinline__ void async_g2l_b32(unsigned lds_byte_addr,
                                              unsigned global_byte_off,
                                              const void* saddr) {
  asm volatile("global_load_async_to_lds_b32 %0, %1, %2"
               :: "v"(lds_byte_addr), "v"(global_byte_off), "s"(saddr)
               : "memory");
}
__device__ __forceinline__ void wait_async0() {
  asm volatile("s_wait_asynccnt 0x0" ::: "memory");
}
__device__ __forceinline__ unsigned lds_off(const void* p) {
  return (unsigned)(uintptr_t)p;   // low 32 bits of generic LDS addr = offset
}

// 8-point twiddles: cos/sin(2*pi*r/8)
__constant__ float CTAB[8] = {1.f, 0.70710678f, 0.f, -0.70710678f,
                              -1.f, -0.70710678f, 0.f, 0.70710678f};
__constant__ float STAB[8] = {0.f, 0.70710678f, 1.f, 0.70710678f,
                              0.f, -0.70710678f, -1.f, -0.70710678f};

// ---------------------------------------------------------------- kv GEMM
__global__ __launch_bounds__(128) void fsas_kv(const float* __restrict__ prior,
                                               const float* __restrict__ w_kernel,
                                               float* __restrict__ kv) {
  const int b = blockIdx.x;
  const int k = threadIdx.x;      // 0..127
  float s = 0.f;
  #pragma unroll 4
  for (int f = 0; f < 192; ++f)
    s += prior[b * 192 + f] * w_kernel[k * 192 + f];
  kv[b * 128 + k] = s;
}

// ---------------------- fused scale/shift + 1x1 conv (WMMA) + depthwise 3x3
__global__ __launch_bounds__(256) void fsas_gemm_dw(
    const float* __restrict__ x, const float* __restrict__ kv,
    const float* __restrict__ w_hidden, const float* __restrict__ w_dw,
    float* __restrict__ qkv) {
  __shared__ float xs[CIN * NPOSP];    // [64][336] xhat halo tile (padded, zeroed)
  __shared__ float hs[CHUNK * NPOSP];  // [16][336] hidden chunk
  __shared__ float wh[CHUNK * 65];     // [16][65]  w_hidden chunk (padded)
  __shared__ float wd[CHUNK * 18];     // [16][18]  w_dw chunk
  __shared__ float kvs[128];

  const int b   = blockIdx.z;
  const int ty0 = blockIdx.y * TILE;
  const int tx0 = blockIdx.x * TILE;
  const int t    = threadIdx.x;
  const int lane = t & 31;
  const int wave = t >> 5;
  const int half = (lane >> 4) & 1;
  const int l16  = lane & 15;

  if (t < 128) kvs[t] = kv[b * 128 + t];
  __syncthreads();

  // load x halo, apply per-channel scale/shift; zero outside image and in the
  // N-padding columns (SAME conv => hidden is zero-padded; GEMM of 0 gives 0)
  for (int idx = t; idx < CIN * NPOSP; idx += 256) {
    int c = idx / NPOSP;
    int p = idx - c * NPOSP;
    float v = 0.0f;
    if (p < NPOS) {
      int gy = ty0 - 1 + p / HALO;
      int gx = tx0 - 1 + p % HALO;
      if (gy >= 0 && gy < HW && gx >= 0 && gx < HW)
        v = x[((size_t)(b * CIN + c) << 16) + gy * HW + gx] * kvs[c] + kvs[64 + c];
    }
    xs[idx] = v;
  }

  for (int c0 = 0; c0 < QKVC; c0 += CHUNK) {
    __syncthreads();
    for (int i = t; i < CHUNK * 64; i += 256) {
      int m = i >> 6, k = i & 63;
      wh[m * 65 + k] = w_hidden[(c0 + m) * 64 + k];
    }
    if (t < CHUNK * 18) wd[t] = w_dw[c0 * 18 + t];
    __syncthreads();

    // hidden[16][336] = wh[16][64] @ xs[64][336]  via WMMA f32 16x16x4
    for (int nt = wave; nt < NT_A; nt += 8) {   // wave-uniform
      const int n = nt * 16 + l16;
      v8f acc = {0.f, 0.f, 0.f, 0.f, 0.f, 0.f, 0.f, 0.f};
      #pragma unroll
      for (int ks = 0; ks < 16; ++ks) {
        const int kk = 4 * ks + 2 * half;       // ISA A/B lane layout
        v2f a, bb;
        a[0]  = wh[l16 * 65 + kk];
        a[1]  = wh[l16 * 65 + kk + 1];
        bb[0] = xs[kk * NPOSP + n];
        bb[1] = xs[(kk + 1) * NPOSP + n];
        acc = wmma4(a, bb, acc);
      }
      #pragma unroll
      for (int r = 0; r < 8; ++r)               // D: m = r + 8*half
        hs[(r + 8 * half) * NPOSP + n] = acc[r];
    }
    __syncthreads();

    // grouped depthwise 3x3: out chan (c0+oc) uses hidden chans (oc&~1, +1)
    {
      const int oy = t >> 4, ox = t & 15;
      #pragma unroll
      for (int oc = 0; oc < CHUNK; ++oc) {
        const int i0 = oc & ~1;
        float acc = 0.0f;
        #pragma unroll
        for (int ky = 0; ky < 3; ++ky) {
          #pragma unroll
          for (int kx = 0; kx < 3; ++kx) {
            const int p = (oy + ky) * HALO + (ox + kx);
            acc += hs[i0 * NPOSP + p]       * wd[oc * 18 + ky * 3 + kx]
                 + hs[(i0 + 1) * NPOSP + p] * wd[oc * 18 + 9 + ky * 3 + kx];
          }
        }
        qkv[((size_t)(b * QKVC + c0 + oc) << 16) + (ty0 + oy) * HW + (tx0 + ox)] = acc;
      }
    }
  }
}

// ------- per-patch circular conv as DFT GEMMs (WMMA) + LayerNorm(128 chans)
// out = (1/64) * ( AC @ (Qr*Kr - Qi*Ki) + AS @ (Qr*Ki + Qi*Kr) )
// Qr = AC@q, Qi = AS@q, Kr = AC@k, Ki = AS@k
// AC[f][p] = cos(2pi*phi/8), AS[f][p] = -sin(2pi*phi/8),
// phi = (fy*py + fx*px) & 7  -- symmetric, so the same register-resident
// A-fragments serve forward and inverse transforms.
__global__ __launch_bounds__(256) void fsas_fft_ln(
    float* __restrict__ qkv, const float* __restrict__ lnw,
    const float* __restrict__ lnb) {
  __shared__ float qs [64 * PB];   // q staging, later reused for out
  __shared__ float ksb[64 * PB];   // k staging
  __shared__ float orS[64 * PB];   // Re of freq-domain product
  __shared__ float oiS[64 * PB];   // Im of freq-domain product
  __shared__ float ctab[8], stab[8];
  __shared__ float mu[64], rs[64];
  __shared__ float lw[128], lb[128];

  const int b = blockIdx.y;
  const int patch = blockIdx.x;          // 0..1023
  const int py = patch >> 5, px = patch & 31;
  const int t = threadIdx.x;
  const int lane = t & 31, wave = t >> 5;
  const int half = (lane >> 4) & 1, l16 = lane & 15;
  const size_t base = (size_t)b * QKVC * HW2;
  const int porg = (py * 8) * HW + px * 8;

  if (t < 8)   { ctab[t] = CTAB[t]; stab[t] = STAB[t]; }
  if (t < 128) { lw[t] = lnw[t]; lb[t] = lnb[t]; }
  // async global->LDS staging of q and k (transposed+padded scatter)
  for (int idx = t; idx < 8192; idx += 256) {
    int c = idx >> 6, p = idx & 63;
    unsigned g = (unsigned)((base + ((size_t)c << 16) + porg +
                             (p >> 3) * HW + (p & 7)) * sizeof(float));
    async_g2l_b32(lds_off(&qs [p * PB + c]), g, qkv);
    async_g2l_b32(lds_off(&ksb[p * PB + c]), g + (unsigned)(128u * HW2 * 4u), qkv);
  }
  wait_async0();
  __syncthreads();

  const int mt  = wave >> 1;             // each wave owns one 16-row tile
  const int ntb = (wave & 1) * 4;        // and four 16-col tiles
  const int f  = mt * 16 + l16;
  const int fy = f >> 3, fx = f & 7;

  v2f AC[16], AS[16];                    // per-wave A-fragments, registers
  #pragma unroll
  for (int ks = 0; ks < 16; ++ks) {
    const int kkk = 4 * ks + 2 * half;
    #pragma unroll
    for (int v = 0; v < 2; ++v) {
      int p  = kkk + v;
      int ph = (fy * (p >> 3) + fx * (p & 7)) & 7;
      AC[ks][v] = ctab[ph];
      AS[ks][v] = -stab[ph];
    }
  }

  // forward DFTs + pointwise complex product (4 independent WMMA chains)
  for (int nt = ntb; nt < ntb + 4; ++nt) {
    const int n = nt * 16 + l16;
    v8f qr = {0.f,0.f,0.f,0.f,0.f,0.f,0.f,0.f};
    v8f qi = {0.f,0.f,0.f,0.f,0.f,0.f,0.f,0.f};
    v8f kr = {0.f,0.f,0.f,0.f,0.f,0.f,0.f,0.f};
    v8f ki = {0.f,0.f,0.f,0.f,0.f,0.f,0.f,0.f};
    #pragma unroll
    for (int ks = 0; ks < 16; ++ks) {
      const int kkk = 4 * ks + 2 * half;
      v2f bq, bk;
      bq[0] = qs [kkk * PB + n];       bq[1] = qs [(kkk + 1) * PB + n];
      bk[0] = ksb[kkk * PB + n];       bk[1] = ksb[(kkk + 1) * PB + n];
      qr = wmma4(AC[ks], bq, qr);
      qi = wmma4(AS[ks], bq, qi);
      kr = wmma4(AC[ks], bk, kr);
      ki = wmma4(AS[ks], bk, ki);
    }
    v8f orv = qr * kr - qi * ki;
    v8f oiv = qr * ki + qi * kr;
    #pragma unroll
    for (int r = 0; r < 8; ++r) {
      int fr = mt * 16 + r + 8 * half;
      orS[fr * PB + n] = orv[r];
      oiS[fr * PB + n] = oiv[r];
    }
  }
  __syncthreads();

  // inverse transform; two independent accumulation chains, reuse qs as out
  for (int nt = ntb; nt < ntb + 4; ++nt) {
    const int n = nt * 16 + l16;
    v8f acc1 = {0.f,0.f,0.f,0.f,0.f,0.f,0.f,0.f};
    v8f acc2 = {0.f,0.f,0.f,0.f,0.f,0.f,0.f,0.f};
    #pragma unroll
    for (int ks = 0; ks < 16; ++ks) {
      const int kkk = 4 * ks + 2 * half;
      v2f br, bi;
      br[0] = orS[kkk * PB + n];       br[1] = orS[(kkk + 1) * PB + n];
      bi[0] = oiS[kkk * PB + n];       bi[1] = oiS[(kkk + 1) * PB + n];
      acc1 = wmma4(AC[ks], br, acc1);
      acc2 = wmma4(AS[ks], bi, acc2);
    }
    v8f acc = acc1 + acc2;
    #pragma unroll
    for (int r = 0; r < 8; ++r) {
      int p = mt * 16 + r + 8 * half;
      qs[p * PB + n] = acc[r] * 0.015625f;      // 1/64 IDFT normalization
    }
  }
  __syncthreads();

  if (t < 64) {                           // LN stats per position, 128 chans
    float s = 0.f, s2 = 0.f;
    for (int c = 0; c < 128; ++c) {
      float v = qs[t * PB + c];
      s += v; s2 += v * v;
    }
    float m = s * 0.0078125f;
    float var = s2 * 0.0078125f - m * m;
    mu[t] = m;
    rs[t] = rsqrtf(var + 1e-5f);
  }
  __syncthreads();

  for (int idx = t; idx < 8192; idx += 256) {   // overwrite q region
    int c = idx >> 6, p = idx & 63;
    float v = (qs[p * PB + c] - mu[p]) * rs[p] * lw[c] + lb[c];
    qkv[base + ((size_t)c << 16) + porg + (p >> 3) * HW + (p & 7)] = v;
  }
}

// --------------------------- vout = v * out ; 128 -> 64 projection (WMMA)
__global__ __launch_bounds__(256) void fsas_proj(
    const float* __restrict__ qkv, const float* __restrict__ w_proj,
    float* __restrict__ out) {
  __shared__ float vo [128 * 64];   // [128 k][64 n]   v*out tile
  __shared__ float wpT[128 * 64];   // [128 k][64 row] w_proj transposed
  const int b  = blockIdx.y;
  const int p0 = blockIdx.x * 64;
  const int t = threadIdx.x;
  const int lane = t & 31, wave = t >> 5;
  const int half = (lane >> 4) & 1, l16 = lane & 15;
  const size_t base = (size_t)b * QKVC * HW2;

  // async transpose-stage of w_proj
  for (int idx = t; idx < 8192; idx += 256) {
    int row = idx >> 7, k2 = idx & 127;
    async_g2l_b32(lds_off(&wpT[k2 * 64 + row]),
                  (unsigned)((row * 128 + k2) * sizeof(float)), w_proj);
  }
  // vo = LN_out * v  (fused multiply -> VALU staging path)
  for (int idx = t; idx < 8192; idx += 256) {
    int c = idx >> 6, n = idx & 63;
    float ov = qkv[base + ((size_t)c << 16) + p0 + n];          // LN output
    float vv = qkv[base + ((size_t)(256 + c) << 16) + p0 + n];  // v
    vo[c * 64 + n] = ov * vv;
  }
  wait_async0();
  __syncthreads();

  const int mt = wave >> 1;            // each wave: one M tile, two N tiles
  #pragma unroll
  for (int r2 = 0; r2 < 2; ++r2) {
    const int nt = ((wave & 1) << 1) | r2;
    v8f acc1 = {0.f, 0.f, 0.f, 0.f, 0.f, 0.f, 0.f, 0.f};
    v8f acc2 = {0.f, 0.f, 0.f, 0.f, 0.f, 0.f, 0.f, 0.f};
    #pragma unroll 4
    for (int ks = 0; ks < 16; ++ks) {        // two independent 16-deep chains
      const int ka = 4 * ks + 2 * half;
      const int kb = ka + 64;
      v2f a1, b1, a2, b2;
      a1[0] = wpT[ka * 64 + (mt * 16 + l16)];
      a1[1] = wpT[(ka + 1) * 64 + (mt * 16 + l16)];
      b1[0] = vo[ka * 64 + (nt * 16 + l16)];
      b1[1] = vo[(ka + 1) * 64 + (nt * 16 + l16)];
      a2[0] = wpT[kb * 64 + (mt * 16 + l16)];
      a2[1] = wpT[(kb + 1) * 64 + (mt * 16 + l16)];
      b2[0] = vo[kb * 64 + (nt * 16 + l16)];
      b2[1] = vo[(kb + 1) * 64 + (nt * 16 + l16)];
      acc1 = wmma4(a1, b1, acc1);
      acc2 = wmma4(a2, b2, acc2);
    }
    v8f acc = acc1 + acc2;
    #pragma unroll
    for (int r = 0; r < 8; ++r) {
      int M = mt * 16 + r + 8 * half;
      out[((size_t)(b * 64 + M) << 16) + p0 + nt * 16 + l16] = acc[r];
    }
  }
}

// -------------------------------------------------------------------- host
extern "C" void kernel_launch(void* const* d_in, const int* in_sizes, int n_in,
                              void* d_out, int out_size, void* d_ws, size_t ws_size,
                              hipStream_t stream) {
  const float* x        = (const float*)d_in[0];
  const float* prior    = (const float*)d_in[1];
  const float* w_kernel = (const float*)d_in[2];
  const float* w_hidden = (const float*)d_in[3];
  const float* w_dw     = (const float*)d_in[4];
  const float* w_proj   = (const float*)d_in[5];
  const float* ln_w     = (const float*)d_in[6];
  const float* ln_b     = (const float*)d_in[7];
  float* out = (float*)d_out;

  float* kv  = (float*)d_ws;                       // 4*128 floats
  float* qkv = (float*)((char*)d_ws + 4096);       // 4*384*65536 floats

  fsas_kv<<<dim3(4), dim3(128), 0, stream>>>(prior, w_kernel, kv);
  fsas_gemm_dw<<<dim3(16, 16, 4), dim3(256), 0, stream>>>(
      x, kv, w_hidden, w_dw, qkv);
  fsas_fft_ln<<<dim3(1024, 4), dim3(256), 0, stream>>>(qkv, ln_w, ln_b);
  fsas_proj<<<dim3(1024, 4), dim3(256), 0, stream>>>(qkv, w_proj, out);
}